// MAGNN_metapath_specific_6889127542846
// MI455X (gfx1250) — compile-verified
//
#include <hip/hip_runtime.h>
#include <stdint.h>

namespace {
constexpr int kN  = 100000;   // nodes
constexpr int kE  = 600000;   // edges
constexpr int kH  = 8;        // heads
constexpr int kD  = 16;       // out dim
constexpr int kT  = 20000;    // targets
constexpr int kHD = kH * kD;  // 128 floats per row
}

#if __has_builtin(__builtin_amdgcn_global_load_async_to_lds_b128) && \
    __has_builtin(__builtin_amdgcn_s_wait_asynccnt)
#define MAGNN_ASYNC 1
#else
#define MAGNN_ASYNC 0
#endif

#if MAGNN_ASYNC
typedef int v4i __attribute__((vector_size(16)));
typedef __attribute__((address_space(1))) v4i* gptr_v4i;  // global int4*
typedef __attribute__((address_space(3))) v4i* lptr_v4i;  // LDS int4*
#endif

__device__ __forceinline__ float xor_sum_heads(float v) {
  // stride-4 lane orbit {lane&3 + 4k} visits each head h=0..7 exactly once
  v += __shfl_xor(v, 4, 32);
  v += __shfl_xor(v, 8, 32);
  v += __shfl_xor(v, 16, 32);
  return v;
}

__device__ __forceinline__ float xor_max_heads(float v) {
  v = fmaxf(v, __shfl_xor(v, 4, 32));
  v = fmaxf(v, __shfl_xor(v, 8, 32));
  v = fmaxf(v, __shfl_xor(v, 16, 32));
  return v;
}

// ---------------- workspace zeroing (nft + flags), graph-replay safe -------
__global__ __launch_bounds__(256) void zero_ws_kernel(uint4* __restrict__ p, int n4) {
  const uint4 z = {0u, 0u, 0u, 0u};
  for (int i = blockIdx.x * blockDim.x + threadIdx.x; i < n4; i += gridDim.x * blockDim.x)
    p[i] = z;
}

__global__ __launch_bounds__(256) void flag_kernel(const int* __restrict__ tgt,
                                                   int* __restrict__ flags) {
  int t = blockIdx.x * blockDim.x + threadIdx.x;
  if (t < kT) flags[tgt[t]] = 1;
}

// ---------------- per-edge: dot, head-softmax, write a, gated scatter ------
__device__ __forceinline__ void edge_body(int e, int lane, float4 ef,
                                          const float* __restrict__ node,
                                          const int* __restrict__ dst,
                                          const int* __restrict__ flags,
                                          float* __restrict__ nft,
                                          float* __restrict__ a_out) {
  int sdst = dst[e];                               // wave-uniform
  sdst = __builtin_amdgcn_readfirstlane(sdst);     // force SGPR addressing

  const float4 nd = *(const float4*)(node + (size_t)sdst * kHD + lane * 4);
  float p = ef.x * nd.x + ef.y * nd.y + ef.z * nd.z + ef.w * nd.w;
  p += __shfl_xor(p, 1, 32);                       // reduce D=16 within head
  p += __shfl_xor(p, 2, 32);                       // p = sim[h], h = lane>>2

  float m  = xor_max_heads(p);                     // softmax over heads (H=8)
  float ex = expf(p - m);
  float s  = xor_sum_heads(ex);
  float av = ex / s;

  if ((lane & 3) == 0)                             // one writer per head
    a_out[(size_t)e * kH + (lane >> 2)] = av;

  if (flags[sdst]) {                               // only targeted nodes need nft
    float* base = nft + (size_t)sdst * kHD + lane * 4;
    atomicAdd(base + 0, ef.x * av);
    atomicAdd(base + 1, ef.y * av);
    atomicAdd(base + 2, ef.z * av);
    atomicAdd(base + 3, ef.w * av);
  }
}

__global__ __launch_bounds__(256) void edge_kernel(const float* __restrict__ node,
                                                   const float* __restrict__ eft,
                                                   const int* __restrict__ dst,
                                                   const int* __restrict__ flags,
                                                   float* __restrict__ nft,
                                                   float* __restrict__ a_out) {
  const int lane = threadIdx.x & 31;
  const int wib  = threadIdx.x >> 5;
  const int wpb  = blockDim.x >> 5;
  const int nw   = wpb * gridDim.x;
  const int w    = blockIdx.x * wpb + wib;

#if MAGNN_ASYNC
  // double-buffered async global->LDS stream of eft rows (512 B / edge / wave)
  __shared__ float4 stage[8][2][32];
  if (w < kE) {
    const float4* g0 = (const float4*)eft + (size_t)w * 32 + lane;
    __builtin_amdgcn_global_load_async_to_lds_b128(
        (gptr_v4i)(uintptr_t)g0,
        (lptr_v4i)&stage[wib][0][lane], 0, 0);
  }
  int buf = 0;
  for (int e = w; e < kE; e += nw, buf ^= 1) {
    const int en = e + nw;
    if (en < kE) {
      const float4* gn = (const float4*)eft + (size_t)en * 32 + lane;
      __builtin_amdgcn_global_load_async_to_lds_b128(
          (gptr_v4i)(uintptr_t)gn,
          (lptr_v4i)&stage[wib][buf ^ 1][lane], 0, 0);
      __builtin_amdgcn_s_wait_asynccnt(1);         // current buffer ready
    } else {
      __builtin_amdgcn_s_wait_asynccnt(0);
    }
    asm volatile("" ::: "memory");                 // keep LDS read after the wait
    float4 ef = stage[wib][buf][lane];
    edge_body(e, lane, ef, node, dst, flags, nft, a_out);
  }
#else
  for (int e = w; e < kE; e += nw) {
    const int en = e + nw;
    if (en < kE)
      __builtin_prefetch(eft + (size_t)en * kHD + lane * 4, 0, 0);
    float4 ef = *(const float4*)(eft + (size_t)e * kHD + lane * 4);
    edge_body(e, lane, ef, node, dst, flags, nft, a_out);
  }
#endif
}

// ---------------- per-target: gather, L2-normalize over heads --------------
__global__ __launch_bounds__(256) void target_kernel(const float* __restrict__ nft,
                                                     const int* __restrict__ tgt,
                                                     float* __restrict__ out) {
  const int lane = threadIdx.x & 31;
  const int t = blockIdx.x * (blockDim.x >> 5) + (threadIdx.x >> 5);
  if (t >= kT) return;
  int g = tgt[t];
  g = __builtin_amdgcn_readfirstlane(g);

  float4 x = *(const float4*)(nft + (size_t)g * kHD + lane * 4);
  x.x += 1e-15f; x.y += 1e-15f; x.z += 1e-15f; x.w += 1e-15f;

  // element (lane*4+k): d = 4*(lane&3)+k fixed on the stride-4 orbit, h varies
  float nx = xor_sum_heads(x.x * x.x);
  float ny = xor_sum_heads(x.y * x.y);
  float nz = xor_sum_heads(x.z * x.z);
  float nw = xor_sum_heads(x.w * x.w);

  float4 o;
  o.x = x.x / fmaxf(sqrtf(nx), 1e-12f);
  o.y = x.y / fmaxf(sqrtf(ny), 1e-12f);
  o.z = x.z / fmaxf(sqrtf(nz), 1e-12f);
  o.w = x.w / fmaxf(sqrtf(nw), 1e-12f);
  *(float4*)(out + (size_t)t * kHD + lane * 4) = o;
}

extern "C" void kernel_launch(void* const* d_in, const int* in_sizes, int n_in,
                              void* d_out, int out_size, void* d_ws, size_t ws_size,
                              hipStream_t stream) {
  const float* node = (const float*)d_in[0];   // [N,H,D]
  const float* eft  = (const float*)d_in[1];   // [E,H,D]
  const int*   dst  = (const int*)d_in[2];     // [E]
  const int*   tgt  = (const int*)d_in[3];     // [T]

  float* out   = (float*)d_out;                      // [T,H,D]
  float* a_out = out + (size_t)kT * kHD;             // [E,H]

  float* nft  = (float*)d_ws;                                           // [N,H,D]
  int*  flags = (int*)((char*)d_ws + (size_t)kN * kHD * sizeof(float)); // [N]

  const size_t zero_bytes = (size_t)kN * kHD * sizeof(float) + (size_t)kN * sizeof(int);
  const int n4 = (int)(zero_bytes / 16);             // 51.6 MB, 16B-divisible

  zero_ws_kernel<<<1024, 256, 0, stream>>>((uint4*)d_ws, n4);
  flag_kernel<<<(kT + 255) / 256, 256, 0, stream>>>(tgt, flags);
  edge_kernel<<<4096, 256, 0, stream>>>(node, eft, dst, flags, nft, a_out);
  target_kernel<<<(kT + 7) / 8, 256, 0, stream>>>(nft, tgt, out);
}